// ImplicitMoE_65369402245525
// MI455X (gfx1250) — compile-verified
//
#include <hip/hip_runtime.h>
#include <cstdint>
#include <cstddef>

typedef __attribute__((ext_vector_type(2))) float v2f;
typedef __attribute__((ext_vector_type(4))) float v4f;
typedef __attribute__((ext_vector_type(8))) float v8f;

#define NEURONS     16384
#define NGROUPS     64
#define PER_GROUP   256
#define TILE_ROWS   16
#define WAVES       2
#define ROW_STRIDE  (PER_GROUP + 4)   // +4 floats pad: conflict-free ds_load_2addr_b64 A-fragments

// One wave32 processes one tile = 16 batch rows x one 256-neuron group.
//   1) async-stage the 16KB tile global->LDS (ASYNCcnt path, NT hint, no VGPR transit)
//   2) row sums via V_WMMA_F32_16X16X4_F32 with B == ones (matrix pipe does the reduce)
//   3) elementwise (sum - x) * strength from LDS, coalesced non-temporal b128 stores
__global__ __launch_bounds__(WAVES * 32)
void lateral_inhibition_wmma(const float* __restrict__ spk,
                             const float* __restrict__ strength_p,
                             float* __restrict__ out,
                             int B)
{
    __shared__ __align__(16) float tile_s[WAVES][TILE_ROWS * ROW_STRIDE];
    __shared__ __align__(16) float sums_s[WAVES][TILE_ROWS];

    const int lane = threadIdx.x & 31;
    const int wv   = threadIdx.x >> 5;

    const int tile_id = blockIdx.x * WAVES + wv;
    const int row0    = (tile_id / NGROUPS) * TILE_ROWS;
    const int gcol    = (tile_id % NGROUPS) * PER_GROUP;
    if (row0 >= B) return;

    const float strength = strength_p[0];

    float* lt = &tile_s[wv][0];
    // low 32 bits of the generic LDS address == wave-relative LDS byte offset
    const unsigned lds_base = (unsigned)(uintptr_t)lt;

    // ---------------- async global -> LDS tile staging ----------------
    // INST_OFFSET is added to BOTH the global and LDS address (ISA 08 §4.4),
    // and within a row both layouts are contiguous, so each computed address
    // pair covers two b128 loads (offset:0 and offset:512 == +128 floats).
    const float* gbase = spk + (size_t)row0 * NEURONS + gcol;
    #pragma unroll
    for (int row = 0; row < TILE_ROWS; ++row) {
        const unsigned lds_addr =
            lds_base + (unsigned)((row * ROW_STRIDE) * 4 + lane * 16);
        const unsigned long long gp =
            (unsigned long long)(uintptr_t)(gbase + (size_t)row * NEURONS + lane * 4);
        asm volatile("global_load_async_to_lds_b128 %0, %1, off th:TH_LOAD_NT"
                     :
                     : "v"(lds_addr), "v"(gp)
                     : "memory");
        asm volatile("global_load_async_to_lds_b128 %0, %1, off offset:512 th:TH_LOAD_NT"
                     :
                     : "v"(lds_addr), "v"(gp)
                     : "memory");
    }
    asm volatile("s_wait_asynccnt 0" ::: "memory");

    // ------------- group row-sums on the matrix pipe -------------
    // A fragment (16x4 f32, ISA layout): lanes 0-15 hold M=lane, K={k,k+1};
    // lanes 16-31 hold M=lane-16, K={k+2,k+3}.  B = all-ones 4x16 (layout moot).
    const int arow = lane & 15;
    const int koff = (lane >> 4) * 2;
    v2f ones; ones[0] = 1.0f; ones[1] = 1.0f;
    v8f acc = {};
    #pragma unroll
    for (int k = 0; k < PER_GROUP / 4; ++k) {
        const v2f a = *(const v2f*)(lt + arow * ROW_STRIDE + (k * 4 + koff));
        acc = __builtin_amdgcn_wmma_f32_16x16x4_f32(
            false, a, false, ones, (short)0, acc, false, false);
    }

    // C/D layout: lane 0 holds sums of rows 0-7 in acc[0..7], lane 16 rows 8-15.
    if ((lane & 15) == 0) {
        float* sp = &sums_s[wv][(lane >> 4) * 8];
        #pragma unroll
        for (int r = 0; r < 8; ++r) sp[r] = acc[r];
    }
    // LDS pipe is in-order within a wave: stores above complete before reads below.

    // ---------- elementwise output, coalesced non-temporal b128 ----------
    float* obase = out + (size_t)row0 * NEURONS + gcol;
    #pragma unroll 4
    for (int j = 0; j < 32; ++j) {
        const int t   = lane + 32 * j;
        const int row = t >> 6;
        const int col = (t & 63) * 4;
        const float s = sums_s[wv][row];
        const v4f x = *(const v4f*)(lt + row * ROW_STRIDE + col);
        v4f o;
        o[0] = (s - x[0]) * strength;
        o[1] = (s - x[1]) * strength;
        o[2] = (s - x[2]) * strength;
        o[3] = (s - x[3]) * strength;
        __builtin_nontemporal_store(o, (v4f*)(obase + (size_t)row * NEURONS + col));
    }
}

extern "C" void kernel_launch(void* const* d_in, const int* in_sizes, int n_in,
                              void* d_out, int out_size, void* d_ws, size_t ws_size,
                              hipStream_t stream) {
    const float* spk      = (const float*)d_in[0];
    const float* strength = (const float*)d_in[1];
    float* out            = (float*)d_out;

    const int B      = in_sizes[0] / NEURONS;          // 4096
    const int tiles  = (B / TILE_ROWS) * NGROUPS;      // 16384
    const int blocks = tiles / WAVES;                  // 8192

    lateral_inhibition_wmma<<<blocks, WAVES * 32, 0, stream>>>(spk, strength, out, B);
}